// EmotionCaps_3453153706016
// MI455X (gfx1250) — compile-verified
//
#include <hip/hip_runtime.h>
#include <math.h>

typedef __attribute__((ext_vector_type(2))) float v2f;
typedef __attribute__((ext_vector_type(8))) float v8f;

#define B_   64
#define N_   1024
#define IN_  64
#define E_   8
#define O_   32
#define R_   256   // E_*O_

// ---------------------------------------------------------------------------
// Wave32 reduction helper
// ---------------------------------------------------------------------------
__device__ __forceinline__ float waveReduceSum(float x) {
#pragma unroll
  for (int off = 16; off > 0; off >>= 1)
    x += __shfl_xor(x, off, 32);
  return x;
}

// ---------------------------------------------------------------------------
// Kernel 1: u_hat[b,n,r] = sum_i W[n,r,i] * u[b,n,i]   (r = e*32+o)
// Per-wave: one n, 32 batch rows (2 M-tiles), 64 columns (4 x 16 cols).
// 8 accumulators; per K-step: 2 A-frag + 4 B-frag loads feed 8 WMMAs.
// A fragment (16x4 fp32): lane<16 -> (M=lane, K=k0/k0+1), lane>=16 -> (M=lane-16, K=k0+2/k0+3)
// B fragment (4x16 fp32): same striping with N=lane&15.
// C/D (16x16 f32): VGPR v, lane l -> row m0+v+8*(l>>4), col l&15.
// ---------------------------------------------------------------------------
__global__ __launch_bounds__(256) void uhat_wmma_kernel(
    const float* __restrict__ u,     // [B, N, IN]
    const float* __restrict__ W,     // [N, R, IN]
    float* __restrict__ u_hat) {     // [B, N, R]
  const int wave = (blockIdx.x * blockDim.x + threadIdx.x) >> 5;
  const int lane = threadIdx.x & 31;

  const int n    = wave >> 3;          // 1024
  const int mgrp = (wave >> 2) & 1;    // 2 (batch groups of 32)
  const int rgrp = wave & 3;           // 4 (column groups of 64)
  const int m0 = mgrp * 32;
  const int r0 = rgrp * 64;

  const int ml   = lane & 15;
  const int koff = (lane >> 4) << 1;

  const float* uA0 = u + ((size_t)(m0 + ml) * N_ + n) * IN_ + koff;
  const float* uA1 = uA0 + (size_t)16 * N_ * IN_;
  const float* wB  = W + ((size_t)n * R_ + r0 + ml) * IN_ + koff;

  v8f acc00 = {}, acc01 = {}, acc02 = {}, acc03 = {};
  v8f acc10 = {}, acc11 = {}, acc12 = {}, acc13 = {};

#pragma unroll 4
  for (int k0 = 0; k0 < IN_; k0 += 4) {
    v2f a0 = *(const v2f*)(uA0 + k0);
    v2f a1 = *(const v2f*)(uA1 + k0);
    v2f b0 = *(const v2f*)(wB + k0);
    v2f b1 = *(const v2f*)(wB + 16 * IN_ + k0);
    v2f b2 = *(const v2f*)(wB + 32 * IN_ + k0);
    v2f b3 = *(const v2f*)(wB + 48 * IN_ + k0);
    acc00 = __builtin_amdgcn_wmma_f32_16x16x4_f32(false, a0, false, b0, (short)0, acc00, false, false);
    acc01 = __builtin_amdgcn_wmma_f32_16x16x4_f32(false, a0, false, b1, (short)0, acc01, false, false);
    acc02 = __builtin_amdgcn_wmma_f32_16x16x4_f32(false, a0, false, b2, (short)0, acc02, false, false);
    acc03 = __builtin_amdgcn_wmma_f32_16x16x4_f32(false, a0, false, b3, (short)0, acc03, false, false);
    acc10 = __builtin_amdgcn_wmma_f32_16x16x4_f32(false, a1, false, b0, (short)0, acc10, false, false);
    acc11 = __builtin_amdgcn_wmma_f32_16x16x4_f32(false, a1, false, b1, (short)0, acc11, false, false);
    acc12 = __builtin_amdgcn_wmma_f32_16x16x4_f32(false, a1, false, b2, (short)0, acc12, false, false);
    acc13 = __builtin_amdgcn_wmma_f32_16x16x4_f32(false, a1, false, b3, (short)0, acc13, false, false);
  }

  const int mrowBase = m0 + ((lane >> 4) << 3);
  const int col = lane & 15;
#pragma unroll
  for (int i = 0; i < 8; i++) {
    size_t ob0 = ((size_t)(mrowBase + i) * N_ + n) * R_ + r0 + col;
    u_hat[ob0 +  0] = acc00[i];
    u_hat[ob0 + 16] = acc01[i];
    u_hat[ob0 + 32] = acc02[i];
    u_hat[ob0 + 48] = acc03[i];
    size_t ob1 = ((size_t)(mrowBase + 16 + i) * N_ + n) * R_ + r0 + col;
    u_hat[ob1 +  0] = acc10[i];
    u_hat[ob1 + 16] = acc11[i];
    u_hat[ob1 + 32] = acc12[i];
    u_hat[ob1 + 48] = acc13[i];
  }
}

// ---------------------------------------------------------------------------
// Zero the routing logits b[B,N,E]
// ---------------------------------------------------------------------------
__global__ void zero_kernel(float* __restrict__ p, int count) {
  int i = blockIdx.x * blockDim.x + threadIdx.x;
  if (i < count) p[i] = 0.0f;
}

// ---------------------------------------------------------------------------
// c[b,n,:] = softmax(blog[b,n,:]) over E=8
// ---------------------------------------------------------------------------
__global__ void softmax_kernel(const float* __restrict__ blog, float* __restrict__ c) {
  int idx = blockIdx.x * blockDim.x + threadIdx.x;   // b*N + n
  if (idx >= B_ * N_) return;
  const float* bp = blog + (size_t)idx * E_;
  float mx = bp[0];
#pragma unroll
  for (int e = 1; e < E_; e++) mx = fmaxf(mx, bp[e]);
  float ex[E_];
  float sum = 0.0f;
#pragma unroll
  for (int e = 0; e < E_; e++) { ex[e] = __expf(bp[e] - mx); sum += ex[e]; }
  float inv = 1.0f / sum;
  float* cp = c + (size_t)idx * E_;
#pragma unroll
  for (int e = 0; e < E_; e++) cp[e] = ex[e] * inv;
}

// ---------------------------------------------------------------------------
// s[b,r] = sum_n c[b,n,e] * u_hat[b,n,r]   (r = e*32+o)
// One block per batch b; threadIdx.x = r/4 (64), threadIdx.y = n-phase (16).
// float4 (b128) streaming of L2-resident u_hat; LDS tree reduce (determin.).
// ---------------------------------------------------------------------------
__global__ __launch_bounds__(1024) void s_sum_kernel(
    const float* __restrict__ c, const float* __restrict__ u_hat,
    float* __restrict__ s) {
  const int b  = blockIdx.x;
  const int t4 = threadIdx.x;        // 0..63  -> r in [4*t4, 4*t4+3]
  const int ns = threadIdx.y;        // 0..15
  const int e  = t4 >> 3;            // (4*t4)>>5
  float4 acc = make_float4(0.f, 0.f, 0.f, 0.f);
  for (int n = ns; n < N_; n += 16) {
    size_t base = (size_t)b * N_ + n;
    float cc = c[base * E_ + e];
    float4 uh = *(const float4*)(u_hat + base * R_ + 4 * t4);
    acc.x += cc * uh.x; acc.y += cc * uh.y;
    acc.z += cc * uh.z; acc.w += cc * uh.w;
  }
  __shared__ float4 red[16][64];
  red[ns][t4] = acc;
  __syncthreads();
  if (ns == 0) {
    float4 tot = red[0][t4];
#pragma unroll
    for (int p = 1; p < 16; p++) {
      float4 q = red[p][t4];
      tot.x += q.x; tot.y += q.y; tot.z += q.z; tot.w += q.w;
    }
    *(float4*)(s + (size_t)b * R_ + 4 * t4) = tot;
  }
}

// ---------------------------------------------------------------------------
// v[b,e,:] = squash(s[b,e,:]) over O=32. One wave32 per (b,e).
// ---------------------------------------------------------------------------
__global__ __launch_bounds__(256) void squash_kernel(
    const float* __restrict__ s, float* __restrict__ vout) {
  const int b = blockIdx.x;
  const int e = threadIdx.x >> 5;
  const int o = threadIdx.x & 31;
  float val = s[((size_t)b * E_ + e) * O_ + o];
  float n2  = waveReduceSum(val * val);
  float nrm = sqrtf(n2);
  float scale = n2 / ((1.0f + n2) * (nrm + 1e-8f));
  vout[((size_t)b * E_ + e) * O_ + o] = scale * val;
}

// ---------------------------------------------------------------------------
// blog[b,n,e] += sum_o u_hat[b,n,e,o] * v[b,e,o]. One wave32 per (b,n,e);
// unique owner per logit -> plain (non-atomic) RMW is deterministic.
// ---------------------------------------------------------------------------
__global__ __launch_bounds__(256) void agree_kernel(
    const float* __restrict__ u_hat, const float* __restrict__ v,
    float* __restrict__ blog) {
  const int wid  = blockIdx.x * 8 + (threadIdx.x >> 5);   // b*N*E waves total
  const int lane = threadIdx.x & 31;
  const int b  = wid >> 13;        // N_*E_ = 8192
  const int ne = wid & 8191;
  const int n  = ne >> 3;
  const int e  = ne & 7;
  size_t base = (size_t)b * N_ + n;
  float p = u_hat[base * R_ + e * O_ + lane] * v[((size_t)b * E_ + e) * O_ + lane];
  float sum = waveReduceSum(p);
  if (lane == 0) blog[base * E_ + e] += sum;
}

// ---------------------------------------------------------------------------
// Launch
// ---------------------------------------------------------------------------
extern "C" void kernel_launch(void* const* d_in, const int* in_sizes, int n_in,
                              void* d_out, int out_size, void* d_ws, size_t ws_size,
                              hipStream_t stream) {
  (void)in_sizes; (void)n_in; (void)out_size; (void)ws_size;
  const float* u = (const float*)d_in[0];   // [B, N, IN]
  const float* W = (const float*)d_in[1];   // [N, E, O, IN] == [N, R, IN]
  float* out = (float*)d_out;               // [B, E, O]

  float* u_hat = (float*)d_ws;                              // B*N*R   = 16,777,216 f
  float* c     = u_hat + (size_t)B_ * N_ * R_;              // B*N*E   =    524,288 f
  float* blog  = c     + (size_t)B_ * N_ * E_;              // B*N*E   =    524,288 f
  float* s     = blog  + (size_t)B_ * N_ * E_;              // B*E*O   =     16,384 f
  float* v     = s     + (size_t)B_ * E_ * O_;              // B*E*O   =     16,384 f

  // 1) u_hat via fp32 WMMA: 1024 n * 2 mgroups * 4 rgroups = 8192 waves
  {
    int waves = N_ * 2 * 4;
    dim3 grid(waves / 8), block(256);
    uhat_wmma_kernel<<<grid, block, 0, stream>>>(u, W, u_hat);
  }

  // 2) b-logits = 0
  {
    int count = B_ * N_ * E_;
    zero_kernel<<<(count + 255) / 256, 256, 0, stream>>>(blog, count);
  }

  // 3) dynamic routing, 3 iterations
  for (int r = 0; r < 3; r++) {
    softmax_kernel<<<(B_ * N_ + 255) / 256, 256, 0, stream>>>(blog, c);
    s_sum_kernel<<<B_, dim3(64, 16), 0, stream>>>(c, u_hat, s);
    squash_kernel<<<B_, 256, 0, stream>>>(s, (r == 2) ? out : v);
    if (r < 2) {
      // B*N*E waves / 8 waves per block
      agree_kernel<<<(B_ * N_ * E_) / 8, 256, 0, stream>>>(u_hat, v, blog);
    }
  }
}